// GNNEmbedding_43989055045633
// MI455X (gfx1250) — compile-verified
//
#include <hip/hip_runtime.h>
#include <hip/hip_bf16.h>

#define NODES      100000
#define MB         64                  // GEMM rows per block
#define NODES_PAD  100032              // 1563 * MB  -> no GEMM bounds checks
#define HID        128

typedef __attribute__((ext_vector_type(2))) float v2f;
typedef __attribute__((ext_vector_type(8))) float v8f;

// ---------------- degree / normalization ----------------

__global__ void k_zero_deg(float* deg) {
    int i = blockIdx.x * blockDim.x + threadIdx.x;
    if (i < NODES_PAD) deg[i] = 0.0f;
}

__global__ void k_count_deg(const int* __restrict__ dst, int E, float* deg) {
    int e = blockIdx.x * blockDim.x + threadIdx.x;
    if (e < E) atomicAdd(&deg[dst[e]], 1.0f);
}

// dinv[i] = rsqrt(deg+1) for real nodes, 0 for pad rows (zeroes padded GEMM inputs)
__global__ void k_make_dinv(float* deg) {
    int i = blockIdx.x * blockDim.x + threadIdx.x;
    if (i < NODES_PAD) deg[i] = (i < NODES) ? rsqrtf(deg[i] + 1.0f) : 0.0f;
}

// ---------------- aggregation (self loop init + edge scatter) ----------------

// agg[n][f] = feat[n][f] * dinv[n]  (float4-wide; i indexes float4 elements; real rows only)
__global__ void k_init_agg(const float4* __restrict__ feat, const float* __restrict__ dinv,
                           float4* __restrict__ agg) {
    int i = blockIdx.x * blockDim.x + threadIdx.x;       // NODES*HID/4 = 3.2M elements
    if (i < NODES * (HID / 4)) {
        int n = i >> 5;                                  // 32 float4 per row
        float s = dinv[n];
        float4 v = feat[i];
        v.x *= s; v.y *= s; v.z *= s; v.w *= s;
        agg[i] = v;
    }
}

// agg[dst][f] += feat[src][f] * dinv[src]
// 8 edges per 256-thread block; 32 lanes/edge; float4 read + 4 f32 atomics per lane (L2-resident)
__global__ void k_scatter(const int* __restrict__ src, const int* __restrict__ dst, int E,
                          const float* __restrict__ feat, const float* __restrict__ dinv,
                          float* __restrict__ agg) {
    int e    = blockIdx.x * 8 + (threadIdx.x >> 5);
    int lane = threadIdx.x & 31;
    if (e < E) {
        int s = src[e];
        int d = dst[e];
        float sc = dinv[s];
        float4 v = *(const float4*)(feat + (long)s * HID + lane * 4);
        float* p = agg + (long)d * HID + lane * 4;
        atomicAdd(p + 0, v.x * sc);
        atomicAdd(p + 1, v.y * sc);
        atomicAdd(p + 2, v.z * sc);
        atomicAdd(p + 3, v.w * sc);
    }
}

// ---------------- fp32 WMMA GEMM: C = act( (dinv .* A) @ W + bias ) ----------------
// A, C: NODES_PAD x 128 (padded -> zero bounds checks), W: 128x128 row-major

template <int RELU>
__global__ __launch_bounds__(256)
void k_gemm(const float* __restrict__ A, const float* __restrict__ dinv,
            const float* __restrict__ W, const float* __restrict__ bias,
            float* __restrict__ C) {
    __shared__ float lA[MB][HID];                       // 32 KB of 320 KB/WGP

    const int r0 = blockIdx.x * MB;
    const int t  = threadIdx.x;

    // Stage A tile into LDS, pre-scaled by dinv[row]; float4-wide (b128 load + ds_store_b128)
    #pragma unroll
    for (int i = 0; i < (MB * HID / 4) / 256; ++i) {     // 8 iterations
        int idx  = t + i * 256;
        int row  = idx >> 5;                             // 32 float4 per row
        int col4 = idx & 31;
        float s  = dinv[r0 + row];
        float4 v = *(const float4*)(A + (long)(r0 + row) * HID + col4 * 4);
        v.x *= s; v.y *= s; v.z *= s; v.w *= s;
        *(float4*)(&lA[row][col4 * 4]) = v;
    }
    __syncthreads();

    const int wave = t >> 5;          // 0..7  -> 16-column strip
    const int lane = t & 31;
    const int half = lane >> 4;       // 0/1
    const int l16  = lane & 15;
    const int n0   = wave * 16;

    v8f acc[4] = {};                  // 4 independent 16x16 accumulator chains (64 rows)

    for (int k0 = 0; k0 < HID; k0 += 4) {
        // B 4x16 tile: v0 = rows (k0 | k0+2), v1 = rows (k0+1 | k0+3), N striped over 16 lanes
        v2f b;
        b.x = W[(k0 + 2 * half)     * HID + n0 + l16];
        b.y = W[(k0 + 2 * half + 1) * HID + n0 + l16];
        #pragma unroll
        for (int mt = 0; mt < 4; ++mt) {
            // A 16x4 tile: lane(l16) = row, halves split K in pairs
            int ar = mt * 16 + l16;
            v2f a;
            a.x = lA[ar][k0 + 2 * half];
            a.y = lA[ar][k0 + 2 * half + 1];
            acc[mt] = __builtin_amdgcn_wmma_f32_16x16x4_f32(
                false, a, false, b, (short)0, acc[mt], false, false);
        }
    }

    const float bv = bias[n0 + l16];
    float* cbase = C + (long)r0 * HID + n0 + l16;

    #pragma unroll
    for (int mt = 0; mt < 4; ++mt) {
        #pragma unroll
        for (int j = 0; j < 8; ++j) {
            float v = acc[mt][j] + bv;
            if (RELU) v = fmaxf(v, 0.0f);
            int row = mt * 16 + j + 8 * half;            // C layout: VGPR j -> M=j (lo) / j+8 (hi)
            cbase[(long)row * HID] = v;
        }
    }
}

// ---------------- final sequence gather (float4-wide) ----------------

__global__ void k_gather(const int* __restrict__ seq, const float4* __restrict__ z,
                         const float4* __restrict__ emb, float4* __restrict__ out, int total4) {
    int i = blockIdx.x * blockDim.x + threadIdx.x;       // total/4 elements
    if (i < total4) {
        int p  = i >> 5;                                 // 32 float4 per row
        int f4 = i & 31;
        int tk = seq[p];
        float4 v;
        if (tk >= 0) v = z[tk * (HID / 4) + f4];
        else         v = emb[(tk + 2 + NODES) * (HID / 4) + f4];  // special rows 100000/100001
        out[i] = v;
    }
}

// ---------------- driver ----------------

extern "C" void kernel_launch(void* const* d_in, const int* in_sizes, int n_in,
                              void* d_out, int out_size, void* d_ws, size_t ws_size,
                              hipStream_t stream) {
    const float* emb = (const float*)d_in[0];   // (100002, 128)
    const float* W0  = (const float*)d_in[1];
    const float* b0  = (const float*)d_in[2];
    const float* W1  = (const float*)d_in[3];
    const float* b1  = (const float*)d_in[4];
    const int*   ei  = (const int*)d_in[5];     // (2, E)
    const int*   seq = (const int*)d_in[6];     // (16, 2048)

    const int E = in_sizes[5] / 2;
    const int* src = ei;
    const int* dst = ei + E;
    const size_t rowbytes = (size_t)NODES_PAD * HID * sizeof(float);
    const int  nout4 = in_sizes[6] * (HID / 4);

    char*  ws   = (char*)d_ws;
    float* dinv = (float*)ws;                                   // NODES_PAD floats
    size_t off  = ((size_t)NODES_PAD * sizeof(float) + 1023) & ~(size_t)1023;
    float* agg  = (float*)(ws + off);                           // NODES_PAD x 128
    float* hbuf = (float*)(ws + off + rowbytes);                // NODES_PAD x 128
    float* out  = (float*)d_out;

    const int nblk = (NODES_PAD + 255) / 256;
    const int eblk = (E + 255) / 256;
    const int fblk = (NODES * (HID / 4) + 255) / 256;
    const int sblk = (E + 7) / 8;
    const int gblk = NODES_PAD / MB;                            // 1563, exact
    const int oblk = (nout4 + 255) / 256;

    // normalization
    k_zero_deg <<<nblk, 256, 0, stream>>>(dinv);
    k_count_deg<<<eblk, 256, 0, stream>>>(dst, E, dinv);
    k_make_dinv<<<nblk, 256, 0, stream>>>(dinv);

    // layer 1: agg1 = selfloop(emb) + scatter(emb) ; h = relu((dinv.*agg1) @ W0 + b0)
    k_init_agg<<<fblk, 256, 0, stream>>>((const float4*)emb, dinv, (float4*)agg);
    k_scatter <<<sblk, 256, 0, stream>>>(src, dst, E, emb, dinv, agg);
    k_gemm<1> <<<gblk, 256, 0, stream>>>(agg, dinv, W0, b0, hbuf);

    // layer 2: agg2 = selfloop(h) + scatter(h) ; z = (dinv.*agg2) @ W1 + b1
    k_init_agg<<<fblk, 256, 0, stream>>>((const float4*)hbuf, dinv, (float4*)agg);
    k_scatter <<<sblk, 256, 0, stream>>>(src, dst, E, hbuf, dinv, agg);
    k_gemm<0> <<<gblk, 256, 0, stream>>>(agg, dinv, W1, b1, hbuf);

    // output gather
    k_gather  <<<oblk, 256, 0, stream>>>(seq, (const float4*)hbuf, (const float4*)emb,
                                         (float4*)out, nout4);
}